// Net_39101382263400
// MI455X (gfx1250) — compile-verified
//
#include <hip/hip_runtime.h>
#include <hip/hip_bf16.h>
#include <math.h>

typedef float v2f __attribute__((ext_vector_type(2)));
typedef float v8f __attribute__((ext_vector_type(8)));

#define F_IN 128
#define CH   32

// LDS strides chosen for conflict-free wave32 access patterns (64 banks):
//  A-tile reads: addr = r*S + k (+2 for hi half). S=132 -> banks k+4r (r=0..15
//  distinct), hi half offset +2 lies in a different mod-4 class -> no conflict.
//  S=36 satisfies the same property for the 32-wide H tiles.
//  Weight reads: rows kk and kk+2 across lane halves. S=40 -> 2S%64 = 16,
//  separating the two 16-lane bank windows.
#define XS_STRIDE 132
#define HS_STRIDE 36
#define WS_STRIDE 40

#define R1 32   // rows per block, layer 1 (64 threads = 2 waves)
#define T1 64
#define R2 64   // rows per block, layers 2/3 (128 threads = 4 waves)
#define T2 128

__device__ __forceinline__ v8f wmma_f32(v2f a, v2f b, v8f c) {
  // D = A(16x4, f32) * B(4x16, f32) + C(16x16, f32)
  return __builtin_amdgcn_wmma_f32_16x16x4_f32(
      /*neg_a=*/false, a, /*neg_b=*/false, b,
      /*c_mod=*/(short)0, c, /*reuse_a=*/false, /*reuse_b=*/false);
}

__device__ __forceinline__ void atomic_add_f32(float* p, float v) {
  __hip_atomic_fetch_add(p, v, __ATOMIC_RELAXED, __HIP_MEMORY_SCOPE_AGENT);
}

// ---------------------------------------------------------------------------
// Layer 1: Y = x @ W1 (N x 32), AGG = x @ W2 + b (N x 32).   K = 128.
// ---------------------------------------------------------------------------
__global__ __launch_bounds__(T1) void gemm_layer1(
    const float* __restrict__ x, const float* __restrict__ W1,
    const float* __restrict__ W2, const float* __restrict__ b,
    float* __restrict__ Y, float* __restrict__ AGG, int n_nodes)
{
  __shared__ float xs[R1 * XS_STRIDE];
  __shared__ float w1s[F_IN * WS_STRIDE];
  __shared__ float w2s[F_IN * WS_STRIDE];
  __shared__ float bs[CH];

  const int tid = threadIdx.x;
  const int rowBase = blockIdx.x * R1;

  // Stage x tile (32 x 128) coalesced, float4 granularity.
  for (int i = tid; i < R1 * (F_IN / 4); i += T1) {
    int r  = i / (F_IN / 4);
    int c4 = (i % (F_IN / 4)) * 4;
    int gr = rowBase + r; if (gr >= n_nodes) gr = n_nodes - 1;
    float4 v = *(const float4*)(x + (size_t)gr * F_IN + c4);
    *(float4*)&xs[r * XS_STRIDE + c4] = v;
  }
  // Stage both weight matrices (128 x 32).
  for (int i = tid; i < F_IN * CH; i += T1) {
    int k = i / CH, c = i % CH;
    w1s[k * WS_STRIDE + c] = W1[i];
    w2s[k * WS_STRIDE + c] = W2[i];
  }
  if (tid < CH) bs[tid] = b[tid];
  __syncthreads();

  const int wave = tid >> 5;
  const int lane = tid & 31;
  const int hi   = lane >> 4;        // lane half selects K pair / M+8
  const int lo   = lane & 15;
  const int arow = (wave << 4) + lo; // A-fragment row within tile
  const int kb   = hi << 1;          // K offset within 4-wide step

  v8f y0 = {}, y1 = {}, z0 = {}, z1 = {};
  for (int kk = 0; kk < F_IN; kk += 4) {
    v2f a;
    a.x = xs[arow * XS_STRIDE + kk + kb];
    a.y = xs[arow * XS_STRIDE + kk + kb + 1];
    v2f b10, b11, b20, b21;
    b10.x = w1s[(kk + kb) * WS_STRIDE + lo];
    b10.y = w1s[(kk + kb + 1) * WS_STRIDE + lo];
    b11.x = w1s[(kk + kb) * WS_STRIDE + 16 + lo];
    b11.y = w1s[(kk + kb + 1) * WS_STRIDE + 16 + lo];
    b20.x = w2s[(kk + kb) * WS_STRIDE + lo];
    b20.y = w2s[(kk + kb + 1) * WS_STRIDE + lo];
    b21.x = w2s[(kk + kb) * WS_STRIDE + 16 + lo];
    b21.y = w2s[(kk + kb + 1) * WS_STRIDE + 16 + lo];
    y0 = wmma_f32(a, b10, y0);
    y1 = wmma_f32(a, b11, y1);
    z0 = wmma_f32(a, b20, z0);
    z1 = wmma_f32(a, b21, z1);
  }

  const float bc0 = bs[lo], bc1 = bs[16 + lo];
#pragma unroll
  for (int i = 0; i < 8; ++i) {
    int gr = rowBase + (wave << 4) + i + (hi << 3); // C/D layout: VGPR i = M=i / M=i+8
    if (gr < n_nodes) {
      size_t base = (size_t)gr * CH;
      Y[base + lo]        = y0[i];
      Y[base + 16 + lo]   = y1[i];
      AGG[base + lo]      = z0[i] + bc0;
      AGG[base + 16 + lo] = z1[i] + bc1;
    }
  }
}

// ---------------------------------------------------------------------------
// Layers 2/3: H = relu(HAGG);  Y = H @ W1;  HAGG = H @ W2 + b  (in place,
// each block owns its 64 rows).   K = 32.
// ---------------------------------------------------------------------------
__global__ __launch_bounds__(T2) void gemm_layer23(
    float* HAGG, const float* __restrict__ W1, const float* __restrict__ W2,
    const float* __restrict__ b, float* __restrict__ Y, int n_nodes)
{
  __shared__ float hs[R2 * HS_STRIDE];
  __shared__ float w1s[CH * WS_STRIDE];
  __shared__ float w2s[CH * WS_STRIDE];
  __shared__ float bs[CH];

  const int tid = threadIdx.x;
  const int rowBase = blockIdx.x * R2;

  // Stage H tile with fused ReLU.
  for (int i = tid; i < R2 * (CH / 4); i += T2) {
    int r  = i / (CH / 4);
    int c4 = (i % (CH / 4)) * 4;
    int gr = rowBase + r; if (gr >= n_nodes) gr = n_nodes - 1;
    float4 v = *(const float4*)(HAGG + (size_t)gr * CH + c4);
    v.x = fmaxf(v.x, 0.f); v.y = fmaxf(v.y, 0.f);
    v.z = fmaxf(v.z, 0.f); v.w = fmaxf(v.w, 0.f);
    *(float4*)&hs[r * HS_STRIDE + c4] = v;
  }
  for (int i = tid; i < CH * CH; i += T2) {
    int k = i / CH, c = i % CH;
    w1s[k * WS_STRIDE + c] = W1[i];
    w2s[k * WS_STRIDE + c] = W2[i];
  }
  if (tid < CH) bs[tid] = b[tid];
  __syncthreads();

  const int wave = tid >> 5;
  const int lane = tid & 31;
  const int hi   = lane >> 4;
  const int lo   = lane & 15;
  const int arow = (wave << 4) + lo;
  const int kb   = hi << 1;

  v8f y0 = {}, y1 = {}, z0 = {}, z1 = {};
#pragma unroll
  for (int kk = 0; kk < CH; kk += 4) {
    v2f a;
    a.x = hs[arow * HS_STRIDE + kk + kb];
    a.y = hs[arow * HS_STRIDE + kk + kb + 1];
    v2f b10, b11, b20, b21;
    b10.x = w1s[(kk + kb) * WS_STRIDE + lo];
    b10.y = w1s[(kk + kb + 1) * WS_STRIDE + lo];
    b11.x = w1s[(kk + kb) * WS_STRIDE + 16 + lo];
    b11.y = w1s[(kk + kb + 1) * WS_STRIDE + 16 + lo];
    b20.x = w2s[(kk + kb) * WS_STRIDE + lo];
    b20.y = w2s[(kk + kb + 1) * WS_STRIDE + lo];
    b21.x = w2s[(kk + kb) * WS_STRIDE + 16 + lo];
    b21.y = w2s[(kk + kb + 1) * WS_STRIDE + 16 + lo];
    y0 = wmma_f32(a, b10, y0);
    y1 = wmma_f32(a, b11, y1);
    z0 = wmma_f32(a, b20, z0);
    z1 = wmma_f32(a, b21, z1);
  }

  const float bc0 = bs[lo], bc1 = bs[16 + lo];
#pragma unroll
  for (int i = 0; i < 8; ++i) {
    int gr = rowBase + (wave << 4) + i + (hi << 3);
    if (gr < n_nodes) {
      size_t base = (size_t)gr * CH;
      Y[base + lo]         = y0[i];
      Y[base + 16 + lo]    = y1[i];
      HAGG[base + lo]      = z0[i] + bc0;
      HAGG[base + 16 + lo] = z1[i] + bc1;
    }
  }
}

// ---------------------------------------------------------------------------
// SpMM scatter: AGG[row] += val * Y[col].  8 lanes per edge, float4 each.
// ---------------------------------------------------------------------------
__global__ void edge_spmm(const int* __restrict__ arow, const int* __restrict__ acol,
                          const float* __restrict__ aval, const float* __restrict__ Y,
                          float* __restrict__ AGG, long long n_work)
{
  long long gid = (long long)blockIdx.x * blockDim.x + threadIdx.x;
  if (gid >= n_work) return;
  int e  = (int)(gid >> 3);
  int c4 = (int)(gid & 7) << 2;
  int r = arow[e];
  int c = acol[e];
  float v = aval[e];
  float4 y = *(const float4*)(Y + (size_t)c * CH + c4);
  float* dst = AGG + (size_t)r * CH + c4;
  atomic_add_f32(dst + 0, v * y.x);
  atomic_add_f32(dst + 1, v * y.y);
  atomic_add_f32(dst + 2, v * y.z);
  atomic_add_f32(dst + 3, v * y.w);
}

__global__ void zero_kernel(float* __restrict__ p, int n) {
  int i = blockIdx.x * blockDim.x + threadIdx.x;
  if (i < n) p[i] = 0.f;
}

// Pool: per-graph sums of relu(AGG) + node counts. 8 lanes per node.
__global__ void pool_kernel(const float* __restrict__ Hpre, const int* __restrict__ seg,
                            float* __restrict__ pooled, float* __restrict__ cnt, int n_nodes)
{
  long long gid = (long long)blockIdx.x * blockDim.x + threadIdx.x;
  long long n_work = (long long)n_nodes * 8;
  if (gid >= n_work) return;
  int node = (int)(gid >> 3);
  int c4   = (int)(gid & 7) << 2;
  int g = seg[node];
  float4 h = *(const float4*)(Hpre + (size_t)node * CH + c4);
  float* dst = pooled + (size_t)g * CH + c4;
  atomic_add_f32(dst + 0, fmaxf(h.x, 0.f));
  atomic_add_f32(dst + 1, fmaxf(h.y, 0.f));
  atomic_add_f32(dst + 2, fmaxf(h.z, 0.f));
  atomic_add_f32(dst + 3, fmaxf(h.w, 0.f));
  if (c4 == 0) atomic_add_f32(cnt + g, 1.0f);
}

// Head: softmax((pooled/cnt) @ Wd + bd).  One thread per graph.
__global__ void head_kernel(const float* __restrict__ pooled, const float* __restrict__ cnt,
                            const float* __restrict__ Wd, const float* __restrict__ bd,
                            float* __restrict__ out, int n_graphs, int n_labels)
{
  int g = blockIdx.x * blockDim.x + threadIdx.x;
  if (g >= n_graphs) return;
  float inv = 1.0f / fmaxf(cnt[g], 1.0f);
  float p[CH];
#pragma unroll
  for (int k = 0; k < CH; ++k) p[k] = pooled[(size_t)g * CH + k] * inv;
  float logits[32];
  float mx = -INFINITY;
  for (int l = 0; l < n_labels; ++l) {
    float s = bd[l];
#pragma unroll
    for (int k = 0; k < CH; ++k) s += p[k] * Wd[k * n_labels + l];
    logits[l] = s;
    mx = fmaxf(mx, s);
  }
  float sum = 0.f;
  for (int l = 0; l < n_labels; ++l) { float e = __expf(logits[l] - mx); logits[l] = e; sum += e; }
  float rs = 1.0f / sum;
  for (int l = 0; l < n_labels; ++l) out[(size_t)g * n_labels + l] = logits[l] * rs;
}

extern "C" void kernel_launch(void* const* d_in, const int* in_sizes, int n_in,
                              void* d_out, int out_size, void* d_ws, size_t ws_size,
                              hipStream_t stream) {
  const float* x     = (const float*)d_in[0];
  const int*   a_row = (const int*)d_in[1];
  const int*   a_col = (const int*)d_in[2];
  const float* a_val = (const float*)d_in[3];
  const int*   seg   = (const int*)d_in[4];
  const float* W1_1  = (const float*)d_in[5];
  const float* W2_1  = (const float*)d_in[6];
  const float* b_1   = (const float*)d_in[7];
  const float* W1_2  = (const float*)d_in[8];
  const float* W2_2  = (const float*)d_in[9];
  const float* b_2   = (const float*)d_in[10];
  const float* W1_3  = (const float*)d_in[11];
  const float* W2_3  = (const float*)d_in[12];
  const float* b_3   = (const float*)d_in[13];
  const float* Wd    = (const float*)d_in[14];
  const float* bd    = (const float*)d_in[15];

  const int N = in_sizes[4];           // seg has one entry per node
  const int E = in_sizes[1];
  const int n_labels = in_sizes[15];
  const int n_graphs = out_size / n_labels;

  float* ws     = (float*)d_ws;
  float* Y      = ws;                                   // [N, 32]
  float* AGG    = Y + (size_t)N * CH;                   // [N, 32]
  float* pooled = AGG + (size_t)N * CH;                 // [G, 32]
  float* cnt    = pooled + (size_t)n_graphs * CH;       // [G]

  const int b1 = (N + R1 - 1) / R1;
  const int b2 = (N + R2 - 1) / R2;
  const long long ework = (long long)E * 8;
  const int eblocks = (int)((ework + 255) / 256);
  const long long pwork = (long long)N * 8;
  const int pblocks = (int)((pwork + 255) / 256);

  // Layer 1
  gemm_layer1<<<b1, T1, 0, stream>>>(x, W1_1, W2_1, b_1, Y, AGG, N);
  edge_spmm<<<eblocks, 256, 0, stream>>>(a_row, a_col, a_val, Y, AGG, ework);
  // Layer 2
  gemm_layer23<<<b2, T2, 0, stream>>>(AGG, W1_2, W2_2, b_2, Y, N);
  edge_spmm<<<eblocks, 256, 0, stream>>>(a_row, a_col, a_val, Y, AGG, ework);
  // Layer 3
  gemm_layer23<<<b2, T2, 0, stream>>>(AGG, W1_3, W2_3, b_3, Y, N);
  edge_spmm<<<eblocks, 256, 0, stream>>>(a_row, a_col, a_val, Y, AGG, ework);
  // Pool + head
  const int zn = n_graphs * CH + n_graphs; // pooled and cnt are contiguous
  zero_kernel<<<(zn + 255) / 256, 256, 0, stream>>>(pooled, zn);
  pool_kernel<<<pblocks, 256, 0, stream>>>(AGG, seg, pooled, cnt, N);
  head_kernel<<<(n_graphs + 255) / 256, 256, 0, stream>>>(
      pooled, cnt, Wd, bd, (float*)d_out, n_graphs, n_labels);
}